// GAT_14276471292554
// MI455X (gfx1250) — compile-verified
//
#include <hip/hip_runtime.h>
#include <hip/hip_bf16.h>

typedef __attribute__((ext_vector_type(2))) float v2f;
typedef __attribute__((ext_vector_type(4))) float v4f;
typedef __attribute__((ext_vector_type(8))) float v8f;

#define NEG_SLOPE 0.2f
#define SM_EPS 1e-16f

__device__ __forceinline__ float wave_reduce_sum(float v) {
#pragma unroll
  for (int off = 16; off > 0; off >>= 1) v += __shfl_down(v, off, 32);
  return v;
}

// Monotonic mapping float -> uint so integer max == float max.
__device__ __forceinline__ unsigned f32_order_key(float x) {
  unsigned u = __float_as_uint(x);
  return (u & 0x80000000u) ? ~u : (u | 0x80000000u);
}
__device__ __forceinline__ float f32_order_inv(unsigned k) {
  unsigned u = (k & 0x80000000u) ? (k & 0x7fffffffu) : ~k;
  return __uint_as_float(u);
}

// ---------------------------------------------------------------------------
// vd[k] = sum_h W[k][h] * att[h]   (folds the W_dst GEMM into a 128-vector)
// ---------------------------------------------------------------------------
__global__ __launch_bounds__(128) void gat_matvec128(
    const float* __restrict__ W, const float* __restrict__ att,
    float* __restrict__ vd) {
  int k = threadIdx.x;
  float s = 0.0f;
#pragma unroll 8
  for (int h = 0; h < 128; ++h) s += W[k * 128 + h] * att[h];
  vd[k] = s;
}

// ---------------------------------------------------------------------------
// Y[nrows,128] = X[nrows,128] @ W[128,128] using V_WMMA_F32_16X16X4_F32.
// Block = 256 threads (8 waves), 128 rows/block, each wave: 16 rows x 128 cols
// (8 accumulator tiles). W^T staged in LDS pre-swizzled into B-fragment
// layout: sB[g=2*kk+half][col][j] = W[4*kk + 2*half + j][col]  (64 KB exact).
// A fragments read directly from global (each element exactly once/block).
// ---------------------------------------------------------------------------
__global__ __launch_bounds__(256) void gat_gemm128_wmma(
    const float* __restrict__ X, const float* __restrict__ W,
    float* __restrict__ Y, int nrows) {
  __shared__ float sB[64 * 128 * 2];  // 64 KB

  for (int idx = threadIdx.x; idx < 128 * 128; idx += 256) {
    int k = idx >> 7, col = idx & 127;
    int g = ((k >> 2) << 1) | ((k >> 1) & 1);
    sB[(((g << 7) | col) << 1) | (k & 1)] = W[idx];
  }
  __syncthreads();

  const int wave = threadIdx.x >> 5;
  const int lane = threadIdx.x & 31;
  const int half = lane >> 4;   // lanes 16-31 hold the K+2 / M+8 halves
  const int rl   = lane & 15;
  const int rowBase = blockIdx.x * 128 + wave * 16;

  // A fragments: lane holds row (rl), K pair {4kk+2*half, 4kk+2*half+1}
  int arow = rowBase + rl;
  if (arow > nrows - 1) arow = nrows - 1;  // clamp for tail block
  const float* xr = X + (size_t)arow * 128 + (half << 1);

  v2f afrag[32];
#pragma unroll
  for (int kk = 0; kk < 32; ++kk) afrag[kk] = *(const v2f*)(xr + (kk << 2));

  v8f acc[8];
#pragma unroll
  for (int ct = 0; ct < 8; ++ct) acc[ct] = (v8f)(0.0f);

  const v2f* sB2 = (const v2f*)sB;
#pragma unroll
  for (int kk = 0; kk < 32; ++kk) {
    const v2f a = afrag[kk];
    const int gbase = ((kk << 1) | half) << 7;
#pragma unroll
    for (int ct = 0; ct < 8; ++ct) {
      v2f b = sB2[gbase + (ct << 4) + rl];
      acc[ct] = __builtin_amdgcn_wmma_f32_16x16x4_f32(
          false, a, false, b, (short)0, acc[ct], false, false);
    }
  }

  // C/D layout: VGPR v -> row v (lanes 0-15) / row v+8 (lanes 16-31), col=rl
#pragma unroll
  for (int ct = 0; ct < 8; ++ct) {
#pragma unroll
    for (int v = 0; v < 8; ++v) {
      int r = rowBase + (half << 3) + v;
      if (r < nrows) Y[(size_t)r * 128 + (ct << 4) + rl] = acc[ct][v];
    }
  }
}

// ---------------------------------------------------------------------------
// Per-node: a_s[n] = <xs[n,:], att_src>, a_d[n] = <xin[n,:], vd>.
// Also zero-initializes the segment max (ordered-key 0 == -inf) and denom.
// One wave per node.
// ---------------------------------------------------------------------------
__global__ __launch_bounds__(256) void gat_scores(
    const float* __restrict__ xs, const float* __restrict__ xin,
    const float* __restrict__ atts, const float* __restrict__ vd,
    float* __restrict__ a_s, float* __restrict__ a_d,
    unsigned* __restrict__ mbits, float* __restrict__ denom, int n) {
  int node = (blockIdx.x * 256 + threadIdx.x) >> 5;
  int lane = threadIdx.x & 31;
  if (node >= n) return;
  size_t base = (size_t)node * 128 + lane * 4;
  v4f xsv = *(const v4f*)(xs + base);
  v4f av  = *(const v4f*)(atts + lane * 4);
  v4f xiv = *(const v4f*)(xin + base);
  v4f vdv = *(const v4f*)(vd + lane * 4);
  float s = xsv.x * av.x + xsv.y * av.y + xsv.z * av.z + xsv.w * av.w;
  float t = xiv.x * vdv.x + xiv.y * vdv.y + xiv.z * vdv.z + xiv.w * vdv.w;
  s = wave_reduce_sum(s);
  t = wave_reduce_sum(t);
  if (lane == 0) {
    a_s[node] = s;
    a_d[node] = t;
    mbits[node] = 0u;   // key-space "-inf"
    denom[node] = 0.0f;
  }
}

// Pass 1: segment max of e over dst (ordered uint atomicMax).
__global__ __launch_bounds__(256) void gat_edge_max(
    const int* __restrict__ src, const int* __restrict__ dst,
    const float* __restrict__ a_s, const float* __restrict__ a_d,
    unsigned* __restrict__ mbits, int nedges) {
  int i = blockIdx.x * 256 + threadIdx.x;
  if (i >= nedges) return;
  float e = a_s[src[i]] + a_d[dst[i]];
  e = (e > 0.0f) ? e : NEG_SLOPE * e;
  atomicMax(mbits + dst[i], f32_order_key(e));
}

// Decode segment max in place (uint key -> float; untouched node -> 0).
__global__ __launch_bounds__(256) void gat_node_m(unsigned* __restrict__ mbits,
                                                  int n) {
  int i = blockIdx.x * 256 + threadIdx.x;
  if (i >= n) return;
  unsigned k = mbits[i];
  ((float*)mbits)[i] = (k == 0u) ? 0.0f : f32_order_inv(k);
}

// Pass 2: ex = exp(e - m[dst]); cache ex; denom[dst] += ex.
__global__ __launch_bounds__(256) void gat_edge_exp(
    const int* __restrict__ src, const int* __restrict__ dst,
    const float* __restrict__ a_s, const float* __restrict__ a_d,
    const float* __restrict__ m, float* __restrict__ exb,
    float* __restrict__ denom, int nedges) {
  int i = blockIdx.x * 256 + threadIdx.x;
  if (i >= nedges) return;
  int d = dst[i];
  float e = a_s[src[i]] + a_d[d];
  e = (e > 0.0f) ? e : NEG_SLOPE * e;
  float ex = __expf(e - m[d]);
  exb[i] = ex;
  unsafeAtomicAdd(denom + d, ex);
}

__global__ __launch_bounds__(256) void gat_zero4(float* __restrict__ p,
                                                 int count4) {
  int i = blockIdx.x * 256 + threadIdx.x;
  if (i < count4) ((v4f*)p)[i] = (v4f)(0.0f);
}

// Pass 3 (the heavy one): out[dst,:] += alpha * xs[src,:].
// One wave per edge, each lane handles 4 channels; f32 atomics hit L2.
__global__ __launch_bounds__(256) void gat_edge_aggregate(
    const int* __restrict__ src, const int* __restrict__ dst,
    const float* __restrict__ exb, const float* __restrict__ denom,
    const float* __restrict__ xs, float* __restrict__ out, int nedges) {
  int e = (blockIdx.x * 256 + threadIdx.x) >> 5;
  if (e >= nedges) return;
  int lane = threadIdx.x & 31;
  int s = src[e], d = dst[e];
  float alpha = exb[e] / (denom[d] + SM_EPS);
  v4f v = *(const v4f*)(xs + (size_t)s * 128 + lane * 4);
  float* o = out + (size_t)d * 128 + lane * 4;
  unsafeAtomicAdd(o + 0, alpha * v.x);
  unsafeAtomicAdd(o + 1, alpha * v.y);
  unsafeAtomicAdd(o + 2, alpha * v.z);
  unsafeAtomicAdd(o + 3, alpha * v.w);
}

__global__ __launch_bounds__(256) void gat_bias_relu(
    float* __restrict__ buf, const float* __restrict__ bias, int n) {
  int i = blockIdx.x * 256 + threadIdx.x;
  if (i >= n * 32) return;
  int node = i >> 5, c4 = i & 31;
  v4f v = *(v4f*)(buf + (size_t)node * 128 + c4 * 4);
  v4f b = *(const v4f*)(bias + c4 * 4);
  v.x = fmaxf(v.x + b.x, 0.0f);
  v.y = fmaxf(v.y + b.y, 0.0f);
  v.z = fmaxf(v.z + b.z, 0.0f);
  v.w = fmaxf(v.w + b.w, 0.0f);
  *(v4f*)(buf + (size_t)node * 128 + c4 * 4) = v;
}

// out[n,0:2] = h[n,:] @ Wl + bl. One wave per node; Wl is [128][2] row-major.
__global__ __launch_bounds__(256) void gat_final_linear(
    const float* __restrict__ h, const float* __restrict__ Wl,
    const float* __restrict__ bl, float* __restrict__ out, int n) {
  int node = (blockIdx.x * 256 + threadIdx.x) >> 5;
  int lane = threadIdx.x & 31;
  if (node >= n) return;
  v4f hv = *(const v4f*)(h + (size_t)node * 128 + lane * 4);
  v4f w0 = *(const v4f*)(Wl + lane * 8);      // k0o0 k0o1 k1o0 k1o1
  v4f w1 = *(const v4f*)(Wl + lane * 8 + 4);  // k2o0 k2o1 k3o0 k3o1
  float p0 = hv.x * w0.x + hv.y * w0.z + hv.z * w1.x + hv.w * w1.z;
  float p1 = hv.x * w0.y + hv.y * w0.w + hv.z * w1.y + hv.w * w1.w;
  p0 = wave_reduce_sum(p0);
  p1 = wave_reduce_sum(p1);
  if (lane == 0) {
    out[(size_t)node * 2 + 0] = p0 + bl[0];
    out[(size_t)node * 2 + 1] = p1 + bl[1];
  }
}

// ---------------------------------------------------------------------------
// Host-side layer driver (all launches on `stream`).
// NOTE: outbuf may alias xin — xin is last consumed by gat_scores, and
// outbuf is first written by gat_zero4 which is stream-ordered after it.
// ---------------------------------------------------------------------------
static void run_gat_layer(const float* xin, const float* Wsrc,
                          const float* Wdst, const float* atts,
                          const float* attd, const float* bias, const int* src,
                          const int* dst, int N, int E, float* xs,
                          float* outbuf, float* a_s, float* a_d, unsigned* mb,
                          float* denom, float* exb, float* vd,
                          hipStream_t stream) {
  gat_matvec128<<<1, 128, 0, stream>>>(Wdst, attd, vd);
  gat_gemm128_wmma<<<(N + 127) / 128, 256, 0, stream>>>(xin, Wsrc, xs, N);
  gat_scores<<<(N + 7) / 8, 256, 0, stream>>>(xs, xin, atts, vd, a_s, a_d, mb,
                                              denom, N);
  gat_edge_max<<<(E + 255) / 256, 256, 0, stream>>>(src, dst, a_s, a_d, mb, E);
  gat_node_m<<<(N + 255) / 256, 256, 0, stream>>>(mb, N);
  gat_edge_exp<<<(E + 255) / 256, 256, 0, stream>>>(src, dst, a_s, a_d,
                                                    (const float*)mb, exb,
                                                    denom, E);
  gat_zero4<<<(N * 32 + 255) / 256, 256, 0, stream>>>(outbuf, N * 32);
  gat_edge_aggregate<<<(E + 7) / 8, 256, 0, stream>>>(src, dst, exb, denom, xs,
                                                      outbuf, E);
  gat_bias_relu<<<(N * 32 + 255) / 256, 256, 0, stream>>>(outbuf, bias, N);
}

extern "C" void kernel_launch(void* const* d_in, const int* in_sizes, int n_in,
                              void* d_out, int out_size, void* d_ws,
                              size_t ws_size, hipStream_t stream) {
  const float* x   = (const float*)d_in[0];
  const int* ei    = (const int*)d_in[1];
  const float* W1s = (const float*)d_in[2];
  const float* W1d = (const float*)d_in[3];
  const float* a1s = (const float*)d_in[4];
  const float* a1d = (const float*)d_in[5];
  const float* b1  = (const float*)d_in[6];
  const float* W2s = (const float*)d_in[7];
  const float* W2d = (const float*)d_in[8];
  const float* a2s = (const float*)d_in[9];
  const float* a2d = (const float*)d_in[10];
  const float* b2  = (const float*)d_in[11];
  const float* Wl  = (const float*)d_in[12];
  const float* bl  = (const float*)d_in[13];

  const int N = in_sizes[0] / 128;
  const int E = in_sizes[1] / 2;
  const int* src = ei;
  const int* dst = ei + E;

  // Workspace layout (floats): bufA | bufB | a_s | a_d | mbits | denom | ex | vd
  float* bufA  = (float*)d_ws;
  float* bufB  = bufA + (size_t)N * 128;
  float* a_s   = bufB + (size_t)N * 128;
  float* a_d   = a_s + N;
  unsigned* mb = (unsigned*)(a_d + N);
  float* denom = (float*)mb + N;
  float* exb   = denom + N;
  float* vd    = exb + E;

  // Layer 1: xin = x, xs -> bufA, aggregate -> bufB (becomes h1)
  run_gat_layer(x, W1s, W1d, a1s, a1d, b1, src, dst, N, E, bufA, bufB, a_s,
                a_d, mb, denom, exb, vd, stream);
  // Layer 2: xin = h1 (bufB), xs -> bufA, aggregate -> bufB (becomes h2)
  run_gat_layer(bufB, W2s, W2d, a2s, a2d, b2, src, dst, N, E, bufA, bufB, a_s,
                a_d, mb, denom, exb, vd, stream);
  // Head
  gat_final_linear<<<(N + 7) / 8, 256, 0, stream>>>(bufB, Wl, bl,
                                                    (float*)d_out, N);
}